// DTFOS_38560216383977
// MI455X (gfx1250) — compile-verified
//
#include <hip/hip_runtime.h>

typedef __attribute__((ext_vector_type(2))) float v2f;
typedef __attribute__((ext_vector_type(8))) float v8f;

#define B_  16
#define T_  8192
#define N_  256
#define TM1 (T_ - 1)

static __device__ __forceinline__ v8f wmma_f32(v2f a, v2f b, v8f c) {
  return __builtin_amdgcn_wmma_f32_16x16x4_f32(false, a, false, b, (short)0, c,
                                               false, false);
}

// ---------------------------------------------------------------------------
// Kernel 1: GL fractional-diff weights, per channel (b,i), channel-major out.
// w[0]=1, w[k] = w[k-1] * (k-1-relu(alpha))/k.   w layout: [B][n][T]
// ---------------------------------------------------------------------------
__global__ void k_weights(const float* __restrict__ alpha, float* __restrict__ w) {
  int id = blockIdx.x * blockDim.x + threadIdx.x;   // b*N_ + i
  if (id >= B_ * N_) return;
  float a = alpha[id];
  a = a > 0.f ? a : 0.f;                            // relu
  float* wc = w + (size_t)id * T_;
  float wp = 1.f;
  wc[0] = 1.f;
  for (int k = 1; k < T_; ++k) {
    wp *= ((float)(k - 1) - a) / (float)k;
    wc[k] = wp;
  }
}

// ---------------------------------------------------------------------------
// Kernel 2: transpose X [B][T][n] -> Xt [B][n][T] (LDS 32x33 tile, coalesced)
// ---------------------------------------------------------------------------
__global__ void k_transpose(const float* __restrict__ X, float* __restrict__ Xt) {
  __shared__ float tile[32][33];
  int b = blockIdx.z, t0 = blockIdx.x * 32, i0 = blockIdx.y * 32;
  int tx = threadIdx.x, ty = threadIdx.y;
#pragma unroll
  for (int rr = 0; rr < 4; ++rr) {
    int r = ty + 8 * rr;
    tile[r][tx] = X[((size_t)b * T_ + t0 + r) * N_ + i0 + tx];
  }
  __syncthreads();
#pragma unroll
  for (int rr = 0; rr < 4; ++rr) {
    int r = ty + 8 * rr;
    Xt[((size_t)b * N_ + i0 + r) * T_ + t0 + tx] = tile[tx][r];
  }
}

// ---------------------------------------------------------------------------
// Kernel 3: causal Toeplitz-block convolution on V_WMMA_F32_16X16X4_F32.
// Wave handles (b, i, g); group g = output time-blocks I in [16g, 16g+16).
// For block distance d: D[p,I'] += W_d * Xblk, W_d[p,r] = w[16d + p - r].
// A-frag (16x4): lane row = lane%16; K = {0,1} lanes 0-15, {2,3} lanes 16-31.
// B-frag (4x16): lane col = lane%16; same K striping -> aligned b64 from Xt.
// C/D (16x16):   col = lane%16, row M = vgpr + 8*(lane/16).
// Three phases so the main loop has NO predication:
//   d == 0          : triangular tile (clamped-index load + select)
//   d = 1 .. I0     : fully in-range, unconditional loads
//   d = I0+1..I0+15 : X columns with I < d zero-padded via select
// ---------------------------------------------------------------------------
__global__ void k_conv(const float* __restrict__ Xt, const float* __restrict__ w,
                       float* __restrict__ Y) {
  const int g    = blockIdx.x;               // 0..31
  const int wave = threadIdx.x >> 5;         // 4 waves / block
  const int i    = blockIdx.y * 4 + wave;    // channel
  const int b    = blockIdx.z;               // batch
  const int lane = threadIdx.x & 31;
  const int hi   = lane >> 4;
  const int lo   = lane & 15;

  const float* xc = Xt + ((size_t)b * N_ + i) * T_;
  const float* wc = w  + ((size_t)b * N_ + i) * T_;

  const int I0 = g * 16;
  v8f acc0 = {}, acc1 = {};

  // Loop-invariant per-lane offsets:
  //   A chunk c: K = 4c + 2*hi (x) and +1 (y); w index = 16d + (lo - K)
  //   B chunk c: pair at Xt index 16*(I0 - d) + 16*lo + 2*hi + 4c
  const int woff = lo - 2 * hi;              // w idx = 16d + woff - 4c (a.x)
  const int xoff = 16 * lo + 2 * hi;

  // ---- phase 1: d == 0 (triangular, mask negative lags) ----
  {
    const float* xp = xc + 16 * I0 + xoff;
#pragma unroll
    for (int c = 0; c < 4; ++c) {
      int ix = woff - 4 * c, iy = ix - 1;
      float vx = wc[ix >= 0 ? ix : 0];
      float vy = wc[iy >= 0 ? iy : 0];
      v2f a;
      a.x = ix >= 0 ? vx : 0.f;
      a.y = iy >= 0 ? vy : 0.f;
      v2f bb = *(const v2f*)(xp + 4 * c);
      if (c & 1) acc1 = wmma_f32(a, bb, acc1);
      else       acc0 = wmma_f32(a, bb, acc0);
    }
  }

  // ---- phase 2: d = 1..I0 (hot loop, no predication) ----
  for (int d = 1; d <= I0; ++d) {
    const float* wp = wc + 16 * d + woff;
    const float* xp = xc + 16 * (I0 - d) + xoff;
    __builtin_prefetch(wp + 16, 0, 1);
    __builtin_prefetch(xp - 16, 0, 1);
#pragma unroll
    for (int c = 0; c < 4; ++c) {
      v2f a, bb;
      a.x = wp[-4 * c];
      a.y = wp[-4 * c - 1];
      bb  = *(const v2f*)(xp + 4 * c);
      if (c & 1) acc1 = wmma_f32(a, bb, acc1);
      else       acc0 = wmma_f32(a, bb, acc0);
    }
  }

  // ---- phase 3: d = I0+1 .. I0+15 (zero-pad X where block index < 0) ----
  for (int d = I0 + 1; d <= I0 + 15; ++d) {
    const int sblk = I0 - d + lo;            // source block, may be < 0
    const bool ok  = sblk >= 0;
    const float* wp = wc + 16 * d + woff;
    const float* xp = xc + 16 * (ok ? sblk : 0) + 2 * hi;
#pragma unroll
    for (int c = 0; c < 4; ++c) {
      v2f a, bb, xv;
      a.x = wp[-4 * c];
      a.y = wp[-4 * c - 1];
      xv  = *(const v2f*)(xp + 4 * c);
      bb.x = ok ? xv.x : 0.f;
      bb.y = ok ? xv.y : 0.f;
      if (c & 1) acc1 = wmma_f32(a, bb, acc1);
      else       acc0 = wmma_f32(a, bb, acc0);
    }
  }

  // write Y[b][t][i]: col I' = lo, row p = v + 8*hi, t = 16*(I0+I') + p
#pragma unroll
  for (int v = 0; v < 8; ++v) {
    int t = 16 * (I0 + lo) + v + 8 * hi;
    Y[((size_t)b * T_ + t) * N_ + i] = acc0[v] + acc1[v];
  }
}

// ---------------------------------------------------------------------------
// Kernel 4: E[b,t,m] = Y[b,t+1,m] - sum_k X[b,t,k]*A[b,m,k]
// One wave per 16(t) x 16(m) tile; K=256 in chunks of 4; dual accumulators.
// ---------------------------------------------------------------------------
__global__ void k_combine(const float* __restrict__ X, const float* __restrict__ Amat,
                          const float* __restrict__ Y, float* __restrict__ E) {
  const int wave = threadIdx.x >> 5;
  const int tt   = blockIdx.x * 4 + wave;    // 0..511
  const int mm   = blockIdx.y;               // 0..15
  const int b    = blockIdx.z;
  const int lane = threadIdx.x & 31;
  const int hi   = lane >> 4;
  const int lo   = lane & 15;
  const int t0 = tt * 16, m0 = mm * 16;

  const float* Xrow = X    + ((size_t)b * T_ + t0 + lo) * N_ + 2 * hi;
  const float* Arow = Amat + ((size_t)b * N_ + m0 + lo) * N_ + 2 * hi;

  v8f acc0 = {}, acc1 = {};
  for (int k0 = 0; k0 < N_; k0 += 8) {
    v2f a0 = *(const v2f*)(Xrow + k0);
    v2f b0 = *(const v2f*)(Arow + k0);
    v2f a1 = *(const v2f*)(Xrow + k0 + 4);
    v2f b1 = *(const v2f*)(Arow + k0 + 4);
    acc0 = wmma_f32(a0, b0, acc0);
    acc1 = wmma_f32(a1, b1, acc1);
  }

#pragma unroll
  for (int v = 0; v < 8; ++v) {
    int t = t0 + v + 8 * hi;                 // D row M = v + 8*hi
    if (t < TM1) {
      float y = Y[((size_t)b * T_ + t + 1) * N_ + m0 + lo];
      E[((size_t)b * TM1 + t) * N_ + m0 + lo] = y - (acc0[v] + acc1[v]);
    }
  }
}

// ---------------------------------------------------------------------------
// Launch: inputs {X[B,T,n], A[B,n,n], alpha[B,n]}, output E[B,T-1,n].
// Workspace (floats): [ Xt | w | Y ], each B*n*T = 33.5M floats (402 MB).
// ---------------------------------------------------------------------------
extern "C" void kernel_launch(void* const* d_in, const int* in_sizes, int n_in,
                              void* d_out, int out_size, void* d_ws, size_t ws_size,
                              hipStream_t stream) {
  const float* X     = (const float*)d_in[0];
  const float* A     = (const float*)d_in[1];
  const float* alpha = (const float*)d_in[2];
  float* E = (float*)d_out;

  const size_t chan = (size_t)B_ * N_ * T_;
  float* Xt = (float*)d_ws;
  float* w  = Xt + chan;
  float* Y  = w + chan;

  k_weights  <<<dim3((B_ * N_ + 255) / 256), dim3(256),   0, stream>>>(alpha, w);
  k_transpose<<<dim3(T_ / 32, N_ / 32, B_),  dim3(32, 8), 0, stream>>>(X, Xt);
  k_conv     <<<dim3(32, N_ / 4, B_),        dim3(128),   0, stream>>>(Xt, w, Y);
  k_combine  <<<dim3(T_ / 16 / 4, N_ / 16, B_), dim3(128), 0, stream>>>(X, A, Y, E);
}